// H2_GAT_88098369176181
// MI455X (gfx1250) — compile-verified
//
#include <hip/hip_runtime.h>
#include <hip/hip_bf16.h>
#include <math.h>

// ---------------------------------------------------------------------------
// Hierarchical GNN forward for MI455X (gfx1250, wave32).
// Dense transforms use V_WMMA_F32_16X16X4_F32; edge ops use float atomics.
// ---------------------------------------------------------------------------

#define NEG_SLOPE 0.2f

typedef float v2f __attribute__((ext_vector_type(2)));
typedef float v8f __attribute__((ext_vector_type(8)));

__device__ __forceinline__ void atomicMaxF(float* addr, float v) {
  // Works for mixed-sign values given init of -inf (0xFF800000).
  if (v >= 0.f) atomicMax((int*)addr, __float_as_int(v));
  else          atomicMin((unsigned int*)addr, __float_as_uint(v));
}

__device__ __forceinline__ float lrelu(float v) { return v > 0.f ? v : NEG_SLOPE * v; }

// ---------------------------------------------------------------------------
// OUT[N,64] = epilogue( X[N,FI] @ W[FI,64] )   FI % 4 == 0
// EPI 0: raw    EPI 1: relu(x+bias)    EPI 2: relu(gamma*(x+bias)+beta)
// 4 waves / block; each wave computes a 16-row x 64-col tile with
// 4x V_WMMA_F32_16X16X4_F32 per K-step of 4.
// A frag (16x4 f32): lane m=lane&15, kh=lane>>4 -> a = {A[m][k+2kh], A[m][k+2kh+1]}
// C frag: vgpr j -> row = row0 + j + kh*8, col = ntile*16 + m
// ---------------------------------------------------------------------------
template<int EPI>
__global__ void k_linear64(const float* __restrict__ X, const float* __restrict__ W,
                           const float* __restrict__ bias, const float* __restrict__ gamma,
                           const float* __restrict__ beta, float* __restrict__ OUT,
                           int N, int FI)
{
  const int wave = threadIdx.x >> 5;
  const int lane = threadIdx.x & 31;
  const int m    = lane & 15;
  const int kh   = lane >> 4;
  const int row0 = (blockIdx.x * 4 + wave) * 16;
  const int row  = row0 + m;
  const bool rv  = row < N;
  const long rbase = (long)(rv ? row : 0) * FI;

  v8f acc[4] = {v8f{}, v8f{}, v8f{}, v8f{}};

  for (int k = 0; k < FI; k += 4) {
    const int ka = k + kh * 2;
    float x0 = X[rbase + ka];
    float x1 = X[rbase + ka + 1];
    v2f a;
    a.x = rv ? x0 : 0.f;
    a.y = rv ? x1 : 0.f;
    const float* w0 = W + (long)ka * 64;        // row ka
    const float* w1 = W + (long)(ka + 1) * 64;  // row ka+1
#pragma unroll
    for (int nt = 0; nt < 4; ++nt) {
      v2f b;
      b.x = w0[nt * 16 + m];
      b.y = w1[nt * 16 + m];
      acc[nt] = __builtin_amdgcn_wmma_f32_16x16x4_f32(
          false, a, false, b, (short)0, acc[nt], false, false);
    }
  }

#pragma unroll
  for (int nt = 0; nt < 4; ++nt) {
#pragma unroll
    for (int j = 0; j < 8; ++j) {
      int r = row0 + j + kh * 8;
      if (r < N) {
        int c = nt * 16 + m;
        float v = acc[nt][j];
        if constexpr (EPI == 1) v = fmaxf(v + bias[c], 0.f);
        if constexpr (EPI == 2) v = fmaxf(gamma[c] * (v + bias[c]) + beta[c], 0.f);
        OUT[(long)r * 64 + c] = v;
      }
    }
  }
}

// ---------------------------------------------------------------------------
// elementwise / edge kernels
// ---------------------------------------------------------------------------
__global__ void k_fill(float* p, float v, long n) {
  long i = (long)blockIdx.x * blockDim.x + threadIdx.x;
  if (i < n) p[i] = v;
}

__global__ void k_finite0(float* p, long n) {
  long i = (long)blockIdx.x * blockDim.x + threadIdx.x;
  if (i < n) { float v = p[i]; if (!isfinite(v)) p[i] = 0.f; }
}

// el/er: per (node, head) dot of z row chunk with attention vectors
__global__ void k_el_er(const float* __restrict__ z, const float* __restrict__ al,
                        const float* __restrict__ ar, float* __restrict__ el,
                        float* __restrict__ er, int N) {
  int i = blockIdx.x * blockDim.x + threadIdx.x;
  if (i >= N * 4) return;
  int n = i >> 2, h = i & 3;
  const float* zr = z + (long)n * 64 + h * 16;
  float sl = 0.f, sr = 0.f;
#pragma unroll
  for (int d = 0; d < 16; ++d) { float zv = zr[d]; sl += zv * al[h*16+d]; sr += zv * ar[h*16+d]; }
  el[i] = sl; er[i] = sr;
}

__global__ void k_edge_max(const int* __restrict__ src, const int* __restrict__ dst,
                           const float* __restrict__ el, const float* __restrict__ er,
                           float* __restrict__ mbuf, int E) {
  int i = blockIdx.x * blockDim.x + threadIdx.x;
  if (i >= E * 4) return;
  int e = i >> 2, h = i & 3;
  int s = src[e], d = dst[e];
  atomicMaxF(&mbuf[d * 4 + h], lrelu(el[s * 4 + h] + er[d * 4 + h]));
}

__global__ void k_edge_expsum(const int* __restrict__ src, const int* __restrict__ dst,
                              const float* __restrict__ el, const float* __restrict__ er,
                              const float* __restrict__ mbuf, float* __restrict__ sbuf,
                              float* __restrict__ exbuf, int E) {
  int i = blockIdx.x * blockDim.x + threadIdx.x;
  if (i >= E * 4) return;
  int e = i >> 2, h = i & 3;
  int s = src[e], d = dst[e];
  float v = lrelu(el[s * 4 + h] + er[d * 4 + h]);
  float ex = expf(v - mbuf[d * 4 + h]);
  exbuf[i] = ex;
  atomicAdd(&sbuf[d * 4 + h], ex);
}

__global__ void k_gat_accum(const int* __restrict__ src, const int* __restrict__ dst,
                            const float* __restrict__ z, const float* __restrict__ exbuf,
                            const float* __restrict__ sbuf, float* __restrict__ out, int E) {
  int i = blockIdx.x * blockDim.x + threadIdx.x;
  if (i >= E * 4) return;
  int e = i >> 2, h = i & 3;
  int s = src[e], d = dst[e];
  float w = exbuf[i] / fmaxf(sbuf[d * 4 + h], 1e-9f);
  const float* zr = z + (long)s * 64 + h * 16;
  float* orow = out + (long)d * 64 + h * 16;
#pragma unroll
  for (int t = 0; t < 16; ++t) atomicAdd(&orow[t], zr[t] * w);
}

__global__ void k_bias_relu(float* __restrict__ out, const float* __restrict__ b, long n) {
  long i = (long)blockIdx.x * blockDim.x + threadIdx.x;
  if (i < n) out[i] = fmaxf(out[i] + b[i & 63], 0.f);
}

// GIN neighbor sum: agg[dst] += h[src], one thread per (edge, 16-col chunk)
__global__ void k_edge_sum(const int* __restrict__ src, const int* __restrict__ dst,
                           const float* __restrict__ h, float* __restrict__ agg, int E) {
  int i = blockIdx.x * blockDim.x + threadIdx.x;
  if (i >= E * 4) return;
  int e = i >> 2, q = i & 3;
  int s = src[e], d = dst[e];
  const float* hr = h + (long)s * 64 + q * 16;
  float* ar = agg + (long)d * 64 + q * 16;
#pragma unroll
  for (int t = 0; t < 16; ++t) atomicAdd(&ar[t], hr[t]);
}

__global__ void k_gin_combine(const float* __restrict__ h, float* __restrict__ agg,
                              const float* __restrict__ eps, long n) {
  long i = (long)blockIdx.x * blockDim.x + threadIdx.x;
  if (i < n) agg[i] = (1.f + eps[0]) * h[i] + agg[i];
}

// segment max of [N,64] into hc[:,colOff:colOff+64] (stride 128)
__global__ void k_segmax(const float* __restrict__ x, const int* __restrict__ gidx,
                         float* __restrict__ hc, int colOff, int N) {
  int i = blockIdx.x * blockDim.x + threadIdx.x;
  if (i >= N * 4) return;
  int n = i >> 2, q = i & 3;
  int g = gidx[n];
  const float* xr = x + (long)n * 64 + q * 16;
  float* hr = hc + (long)g * 128 + colOff + q * 16;
#pragma unroll
  for (int t = 0; t < 16; ++t) atomicMaxF(&hr[t], xr[t]);
}

__global__ void k_build_hin(const float* __restrict__ hc, const float* __restrict__ pca,
                            float* __restrict__ hin, int N) {
  int i = blockIdx.x * blockDim.x + threadIdx.x;
  if (i >= N * 160) return;
  int n = i / 160, c = i % 160;
  hin[i] = (c < 128) ? hc[(long)n * 128 + c] : pca[(long)n * 32 + (c - 128)];
}

// readout: sums[g][colOff + c] += x[n][c]  (sums row stride 192)
__global__ void k_accum_feat(const float* __restrict__ x, const int* __restrict__ gid,
                             float* __restrict__ sums, int colOff, int N) {
  int i = blockIdx.x * blockDim.x + threadIdx.x;
  if (i >= N * 4) return;
  int n = i >> 2, q = i & 3;
  int g = gid[n];
  const float* xr = x + (long)n * 64 + q * 16;
  float* sr = sums + (long)g * 192 + colOff + q * 16;
#pragma unroll
  for (int t = 0; t < 16; ++t) atomicAdd(&sr[t], xr[t]);
}

__global__ void k_count(const int* __restrict__ gid, float* __restrict__ cnt, int n) {
  int i = blockIdx.x * blockDim.x + threadIdx.x;
  if (i < n) atomicAdd(&cnt[gid[i]], 1.f);
}

// final MLP: hg[64,192] -> relu(@W1[192,64]+b1) -> @W2[64,1]+b2
__global__ void k_head(const float* __restrict__ sums, const float* __restrict__ cntA,
                       const float* __restrict__ cntC, const float* __restrict__ W1,
                       const float* __restrict__ b1, const float* __restrict__ W2,
                       const float* __restrict__ b2, float* __restrict__ out) {
  __shared__ float hg[64][192];
  __shared__ float t1[64][64];
  int tid = threadIdx.x;
  for (int i = tid; i < 64 * 192; i += 256) {
    int g = i / 192, c = i % 192;
    float cnt = (c < 128) ? cntA[g] : cntC[g];
    hg[g][c] = sums[i] / fmaxf(cnt, 1.f);
  }
  __syncthreads();
  for (int i = tid; i < 64 * 64; i += 256) {
    int g = i / 64, j = i % 64;
    float acc = b1[j];
    for (int k = 0; k < 192; ++k) acc += hg[g][k] * W1[k * 64 + j];
    t1[g][j] = fmaxf(acc, 0.f);
  }
  __syncthreads();
  if (tid < 64) {
    float acc = b2[0];
    for (int j = 0; j < 64; ++j) acc += t1[tid][j] * W2[j];
    out[tid] = acc;
  }
}

// ---------------------------------------------------------------------------
// host-side orchestration
// ---------------------------------------------------------------------------
static inline unsigned gsz(long n, int bs) { return (unsigned)((n + bs - 1) / bs); }

static void run_gat(const float* h, int FI, const int* src, const int* dst, int E, int N,
                    const float* W, const float* al, const float* ar, const float* b,
                    float* z, float* outbuf, float* el, float* er, float* mb, float* sb,
                    float* exb, hipStream_t st) {
  k_linear64<0><<<gsz(N, 64), 128, 0, st>>>(h, W, nullptr, nullptr, nullptr, z, N, FI);
  k_el_er<<<gsz((long)N * 4, 256), 256, 0, st>>>(z, al, ar, el, er, N);
  k_fill<<<gsz((long)N * 4, 256), 256, 0, st>>>(mb, -INFINITY, (long)N * 4);
  k_fill<<<gsz((long)N * 4, 256), 256, 0, st>>>(sb, 0.f, (long)N * 4);
  k_fill<<<gsz((long)N * 64, 256), 256, 0, st>>>(outbuf, 0.f, (long)N * 64);
  k_edge_max<<<gsz((long)E * 4, 256), 256, 0, st>>>(src, dst, el, er, mb, E);
  k_finite0<<<gsz((long)N * 4, 256), 256, 0, st>>>(mb, (long)N * 4);
  k_edge_expsum<<<gsz((long)E * 4, 256), 256, 0, st>>>(src, dst, el, er, mb, sb, exb, E);
  k_gat_accum<<<gsz((long)E * 4, 256), 256, 0, st>>>(src, dst, z, exb, sb, outbuf, E);
  k_bias_relu<<<gsz((long)N * 64, 256), 256, 0, st>>>(outbuf, b, (long)N * 64);
}

static void run_gin(const float* h, const int* src, const int* dst, int E, int N,
                    const float* W1, const float* b1, const float* gamma, const float* beta,
                    const float* W2, const float* b2, const float* eps,
                    float* agg, float* t1, float* outbuf, hipStream_t st) {
  k_fill<<<gsz((long)N * 64, 256), 256, 0, st>>>(agg, 0.f, (long)N * 64);
  k_edge_sum<<<gsz((long)E * 4, 256), 256, 0, st>>>(src, dst, h, agg, E);
  k_gin_combine<<<gsz((long)N * 64, 256), 256, 0, st>>>(h, agg, eps, (long)N * 64);
  k_linear64<2><<<gsz(N, 64), 128, 0, st>>>(agg, W1, b1, gamma, beta, t1, N, 64);
  k_linear64<1><<<gsz(N, 64), 128, 0, st>>>(t1, W2, b2, nullptr, nullptr, outbuf, N, 64);
}

extern "C" void kernel_launch(void* const* d_in, const int* in_sizes, int n_in,
                              void* d_out, int out_size, void* d_ws, size_t ws_size,
                              hipStream_t stream) {
  // ---- data inputs -------------------------------------------------------
  const float* atom  = (const float*)d_in[0];
  const int*   srcB[2] = {(const int*)d_in[1], (const int*)d_in[3]};
  const int*   dstB[2] = {(const int*)d_in[2], (const int*)d_in[4]};
  const int*   g1dst = (const int*)d_in[5];
  const float* pca   = (const float*)d_in[6];
  const int*   srcI2 = (const int*)d_in[7];
  const int*   dstI2 = (const int*)d_in[8];
  const int*   gidA  = (const int*)d_in[9];
  const int*   gidC2 = (const int*)d_in[10];

  const int NA  = in_sizes[0] / 16;
  const int EB0 = in_sizes[1];
  const int EB1 = in_sizes[3];
  const int NC2 = in_sizes[6] / 32;
  const int EI2 = in_sizes[7];
  const int EBs[2] = {EB0, EB1};

  // ---- param leaves (jax pytree sorted-key order), base = 11 -------------
  // per bottom t (22 leaves): gat0{W,al,ar,b} gat1{W,al,ar,b}
  //                           gin0{W1,W2,b1,b2,beta,eps,gamma} gin1{...}
  // then h2 (8), then out (4)
  auto P = [&](int idx) { return (const float*)d_in[idx]; };
  const int H2B = 11 + 2 * 22;   // 55
  const int OB  = H2B + 8;       // 63

  // ---- workspace carve-up (floats) ---------------------------------------
  float* ws = (float*)d_ws;
  size_t off = 0;
  auto alloc = [&](size_t n) { float* p = ws + off; off += n; return p; };
  float* bufA  = alloc((size_t)NA * 64);
  float* bufB  = alloc((size_t)NA * 64);
  float* bufC  = alloc((size_t)NA * 64);
  float* hidT0 = alloc((size_t)NA * 64);
  float* hidT1 = alloc((size_t)NA * 64);
  float* elb   = alloc((size_t)NA * 4);
  float* erb   = alloc((size_t)NA * 4);
  float* mb    = alloc((size_t)NA * 4);
  float* sb    = alloc((size_t)NA * 4);
  float* exb   = alloc((size_t)(EB0 > EB1 ? EB0 : EB1) * 4);
  float* hc    = alloc((size_t)NC2 * 128);
  float* hin   = alloc((size_t)NC2 * 160);
  float* c2A   = alloc((size_t)NC2 * 64);
  float* c2B   = alloc((size_t)NC2 * 64);
  float* c2C   = alloc((size_t)NC2 * 64);
  float* sums  = alloc((size_t)64 * 192);
  float* cntA  = alloc(64);
  float* cntC  = alloc(64);
  (void)n_in; (void)out_size; (void)ws_size;

  float* hidT[2] = {hidT0, hidT1};

  // ---- bottom encoders ---------------------------------------------------
  for (int t = 0; t < 2; ++t) {
    const int base = 11 + t * 22;
    const int* src = srcB[t];
    const int* dst = dstB[t];
    const int E = EBs[t];
    // GAT layer 0 (16 -> 64): h=atom -> bufB   (z in bufA)
    run_gat(atom, 16, src, dst, E, NA, P(base+0), P(base+1), P(base+2), P(base+3),
            bufA, bufB, elb, erb, mb, sb, exb, stream);
    // GAT layer 1 (64 -> 64): bufB -> bufC
    run_gat(bufB, 64, src, dst, E, NA, P(base+4), P(base+5), P(base+6), P(base+7),
            bufA, bufC, elb, erb, mb, sb, exb, stream);
    // GIN layer 0: bufC -> bufC   (agg=bufB, t1=bufA)
    run_gin(bufC, src, dst, E, NA, P(base+8), P(base+10), P(base+14), P(base+12),
            P(base+9), P(base+11), P(base+13), bufB, bufA, bufC, stream);
    // GIN layer 1: bufC -> hidT[t]
    run_gin(bufC, src, dst, E, NA, P(base+15), P(base+17), P(base+21), P(base+19),
            P(base+16), P(base+18), P(base+20), bufB, bufA, hidT[t], stream);
  }

  // ---- G1 segment-max pooling + PCA concat -------------------------------
  k_fill<<<gsz((long)NC2 * 128, 256), 256, 0, stream>>>(hc, -INFINITY, (long)NC2 * 128);
  k_segmax<<<gsz((long)NA * 4, 256), 256, 0, stream>>>(hidT0, g1dst, hc, 0, NA);
  k_segmax<<<gsz((long)NA * 4, 256), 256, 0, stream>>>(hidT1, g1dst, hc, 64, NA);
  k_finite0<<<gsz((long)NC2 * 128, 256), 256, 0, stream>>>(hc, (long)NC2 * 128);
  k_build_hin<<<gsz((long)NC2 * 160, 256), 256, 0, stream>>>(hc, pca, hin, NC2);

  // ---- H2 GAT stack ------------------------------------------------------
  run_gat(hin, 160, srcI2, dstI2, EI2, NC2, P(H2B+0), P(H2B+1), P(H2B+2), P(H2B+3),
          c2A, c2B, elb, erb, mb, sb, exb, stream);
  run_gat(c2B, 64, srcI2, dstI2, EI2, NC2, P(H2B+4), P(H2B+5), P(H2B+6), P(H2B+7),
          c2A, c2C, elb, erb, mb, sb, exb, stream);

  // ---- readouts + output MLP --------------------------------------------
  k_fill<<<gsz(64 * 192 + 128, 256), 256, 0, stream>>>(sums, 0.f, 64 * 192 + 128);
  k_count<<<gsz(NA, 256), 256, 0, stream>>>(gidA, cntA, NA);
  k_count<<<gsz(NC2, 256), 256, 0, stream>>>(gidC2, cntC, NC2);
  k_accum_feat<<<gsz((long)NA * 4, 256), 256, 0, stream>>>(hidT0, gidA, sums, 0, NA);
  k_accum_feat<<<gsz((long)NA * 4, 256), 256, 0, stream>>>(hidT1, gidA, sums, 64, NA);
  k_accum_feat<<<gsz((long)NC2 * 4, 256), 256, 0, stream>>>(c2C, gidC2, sums, 128, NC2);
  k_head<<<1, 256, 0, stream>>>(sums, cntA, cntC, P(OB+0), P(OB+2), P(OB+1), P(OB+3),
                                (float*)d_out);
}